// SimpleSparseAutoencoder_63642825392741
// MI455X (gfx1250) — compile-verified
//
#include <hip/hip_runtime.h>
#include <hip/hip_bf16.h>

typedef __attribute__((ext_vector_type(2))) float v2f;
typedef __attribute__((ext_vector_type(8))) float v8f;
typedef __attribute__((ext_vector_type(4))) int   v4i;

typedef __attribute__((address_space(1))) v4i* global_v4i_ptr;
typedef __attribute__((address_space(3))) v4i* lds_v4i_ptr;

#define INPUT_DIM    1024
#define HIDDEN_DIM   512
#define EXPANDED_DIM 8192
#define BATCH        4096
#define TOPK         819

// Block tile geometry: 256 threads = 8 waves, 64(M) x 64(N) tile, BK = 32,
// double-buffered LDS. Rows padded to 34 floats: fragment-read banks are
// (34*l) mod 64, all distinct for l=0..15 (gcd(34,64)=2), so ds_load_b64
// fragment reads are conflict-free.
#define BK      32
#define LDS_PAD 34
#define TILE_F  (64 * LDS_PAD)

// ---------------------------------------------------------------------------
// gfx1250 async global->LDS staging (ASYNCcnt path), with safe fallback.
// Probe-learned signature: (v4i AS1*, v4i AS3*, imm offset, imm cpol).
// ---------------------------------------------------------------------------
#if defined(__has_builtin)
#  if __has_builtin(__builtin_amdgcn_global_load_async_to_lds_b128) && \
      __has_builtin(__builtin_amdgcn_s_wait_asynccnt)
#    define USE_ASYNC_LDS 1
#  endif
#endif
#ifndef USE_ASYNC_LDS
#  define USE_ASYNC_LDS 0
#endif

__device__ __forceinline__ void stage16(const float* __restrict__ g, float* l)
{
#if USE_ASYNC_LDS
    __builtin_amdgcn_global_load_async_to_lds_b128(
        (global_v4i_ptr)(void*)g,
        (lds_v4i_ptr)(void*)l,
        /*imm offset=*/0, /*cpol=*/0);
#else
    const float4 v = *(const float4*)g;
    *(float4*)l = v;
#endif
}

// ---------------------------------------------------------------------------
// LDS-tiled, double-buffered f32 WMMA GEMM:
//   C[m,n] = sum_k A[m,k] * B[n,k]   (A @ B^T)
// A: [M,K] row-major, B: [N,K] row-major, C: [M,N] row-major.
// Each wave computes a 16x32 sub-tile (2 accumulators): every A fragment
// feeds two V_WMMA_F32_16X16X4_F32. Tile k+1's async DMA overlaps tile k's
// WMMA stream; async loads complete in order, so s_wait_asynccnt 4 releases
// exactly the current tile.
// ---------------------------------------------------------------------------
__global__ __launch_bounds__(256) void wmma_gemm_nt(const float* __restrict__ A,
                                                    const float* __restrict__ B,
                                                    float* __restrict__ C,
                                                    int M, int N, int K, int relu)
{
    __shared__ float sA[2][TILE_F];   // 2 x 8.5 KB
    __shared__ float sB[2][TILE_F];   // 2 x 8.5 KB

    const int t    = threadIdx.x;     // 0..255
    const int lane = t & 31;
    const int wave = t >> 5;          // 0..7
    const int wm   = wave & 3;        // 4 sub-tiles along M (16 rows each)
    const int wn   = wave >> 2;       // 2 sub-tiles along N (32 cols each)

    const int mBase = blockIdx.y * 64;
    const int nBase = blockIdx.x * 64;

    // Per-thread staging coordinates: 64x32 tile = 512 16B chunks, 2/thread.
    const int row0 = (t + 0)   >> 3;          // chunks 0..255
    const int c40  = ((t + 0) & 7) * 4;
    const int row1 = (t + 256) >> 3;          // chunks 256..511
    const int c41  = ((t + 256) & 7) * 4;

    const float* __restrict__ aSrc0 = A + (size_t)(mBase + row0) * K + c40;
    const float* __restrict__ aSrc1 = A + (size_t)(mBase + row1) * K + c41;
    const float* __restrict__ bSrc0 = B + (size_t)(nBase + row0) * K + c40;
    const float* __restrict__ bSrc1 = B + (size_t)(nBase + row1) * K + c41;

    const int ld0 = row0 * LDS_PAD + c40;
    const int ld1 = row1 * LDS_PAD + c41;

    // f32 16x4 fragment layout: lanes 0-15 hold K={k,k+1}, lanes 16-31 K={k+2,k+3}
    const int l = lane & 15;
    const int g = (lane >> 4) << 1;

    const int aBase  = (wm * 16 + l) * LDS_PAD + g;
    const int b0Base = (wn * 32 + l) * LDS_PAD + g;
    const int b1Base = (wn * 32 + 16 + l) * LDS_PAD + g;

    v8f c0 = {};
    v8f c1 = {};

    const int ntiles = K / BK;

    // Prologue: stage tile 0 into buffer 0 (4 async b128 per thread).
    stage16(aSrc0, &sA[0][ld0]);
    stage16(aSrc1, &sA[0][ld1]);
    stage16(bSrc0, &sB[0][ld0]);
    stage16(bSrc1, &sB[0][ld1]);

    for (int kt = 0; kt < ntiles; ++kt) {
        const int  cur     = kt & 1;
        const int  nxt     = cur ^ 1;
        const bool hasNext = (kt + 1) < ntiles;

        if (hasNext) {
            // Overlap next tile's DMA with this tile's WMMA stream.
            const int k1 = (kt + 1) * BK;
            stage16(aSrc0 + k1, &sA[nxt][ld0]);
            stage16(aSrc1 + k1, &sA[nxt][ld1]);
            stage16(bSrc0 + k1, &sB[nxt][ld0]);
            stage16(bSrc1 + k1, &sB[nxt][ld1]);
        }

#if USE_ASYNC_LDS
        // Async loads retire in order: leaving the newest 4 in flight means
        // the current tile's 4 are complete.
        if (hasNext) __builtin_amdgcn_s_wait_asynccnt(4);
        else         __builtin_amdgcn_s_wait_asynccnt(0);
#endif
        __syncthreads();   // current tile visible to all waves

        const float* __restrict__ pA = &sA[cur][0];
        const float* __restrict__ pB = &sB[cur][0];
#pragma unroll
        for (int kk = 0; kk < BK; kk += 4) {
            v2f a, b0, b1;
            a.x  = pA[aBase + kk];
            a.y  = pA[aBase + kk + 1];
            b0.x = pB[b0Base + kk];
            b0.y = pB[b0Base + kk + 1];
            b1.x = pB[b1Base + kk];
            b1.y = pB[b1Base + kk + 1];
            c0 = __builtin_amdgcn_wmma_f32_16x16x4_f32(false, a, false, b0,
                                                       (short)0, c0, false, false);
            c1 = __builtin_amdgcn_wmma_f32_16x16x4_f32(false, a, false, b1,
                                                       (short)0, c1, false, false);
        }
        __syncthreads();   // all waves done reading buffer `cur` before restage
    }

    // 32-bit 16x16 C/D layout: VGPR v -> M = v (+8 for lanes 16-31), N = lane%16
    const int n    = lane & 15;
    const int moff = (lane >> 4) * 8;
    const int rowb = mBase + wm * 16 + moff;
    const int col0 = nBase + wn * 32 + n;

    float* __restrict__ crow0 = C + (size_t)rowb * N + col0;
    float* __restrict__ crow1 = crow0 + 16;
#pragma unroll
    for (int v = 0; v < 8; ++v) {
        float v0 = c0[v];
        float v1 = c1[v];
        if (relu) {
            v0 = v0 > 0.f ? v0 : 0.f;
            v1 = v1 > 0.f ? v1 : 0.f;
        }
        crow0[(size_t)v * N] = v0;
        crow1[(size_t)v * N] = v1;
    }
}

// ---------------------------------------------------------------------------
// Per-row top-K threshold: one 256-thread workgroup per row; row staged in
// LDS; 4-pass radix select over the (non-negative, hence order-preserving)
// f32 bit patterns; zero everything below the K-th largest, cap ties.
// ---------------------------------------------------------------------------
__global__ __launch_bounds__(256) void topk_select(float* __restrict__ feat, int kkeep)
{
    __shared__ float    row[EXPANDED_DIM];
    __shared__ unsigned hist[256];
    __shared__ unsigned s_prefix, s_remaining, s_tiecnt;

    const int tid = threadIdx.x;
    float* __restrict__ grow = feat + (size_t)blockIdx.x * EXPANDED_DIM;

    for (int i = tid; i < EXPANDED_DIM; i += 256) row[i] = grow[i];
    if (tid == 0) { s_prefix = 0u; s_remaining = (unsigned)kkeep; s_tiecnt = 0u; }
    __syncthreads();

    for (int pass = 0; pass < 4; ++pass) {
        const int shift = 24 - pass * 8;
        hist[tid] = 0u;
        __syncthreads();
        const unsigned pref = s_prefix;
        for (int i = tid; i < EXPANDED_DIM; i += 256) {
            const unsigned u = __float_as_uint(row[i]);
            const bool match = (pass == 0) || ((u >> (unsigned)(shift + 8)) == pref);
            if (match) atomicAdd(&hist[(u >> shift) & 255u], 1u);
        }
        __syncthreads();
        if (tid == 0) {
            unsigned rem = s_remaining, c = 0u;
            for (int d = 255; d >= 0; --d) {
                c += hist[d];
                if (c >= rem) {
                    s_prefix    = (s_prefix << 8) | (unsigned)d;
                    s_remaining = rem - (c - hist[d]);
                    break;
                }
            }
        }
        __syncthreads();
    }

    const unsigned T   = s_prefix;
    const unsigned rem = s_remaining;
    for (int i = tid; i < EXPANDED_DIM; i += 256) {
        const unsigned u = __float_as_uint(row[i]);
        float v = row[i];
        if (u < T) {
            v = 0.f;
        } else if (u == T) {
            const unsigned slot = atomicAdd(&s_tiecnt, 1u);
            if (slot >= rem) v = 0.f;
        }
        grow[i] = v;
    }
}

extern "C" void kernel_launch(void* const* d_in, const int* in_sizes, int n_in,
                              void* d_out, int out_size, void* d_ws, size_t ws_size,
                              hipStream_t stream)
{
    const float* x     = (const float*)d_in[0];   // [4096,1024]
    const float* W_enc = (const float*)d_in[1];   // [512,1024]
    const float* W_exp = (const float*)d_in[2];   // [8192,512]
    const float* W_dec = (const float*)d_in[3];   // [1024,8192]

    float* out    = (float*)d_out;
    float* recon  = out;                                  // [4096,1024]
    float* feat   = out + (size_t)BATCH * INPUT_DIM;      // [4096,8192]
    float* hidden = (float*)d_ws;                         // [4096,512] scratch

    const dim3 blk(256);

    // hidden = x @ W_enc^T
    wmma_gemm_nt<<<dim3(HIDDEN_DIM / 64, BATCH / 64), blk, 0, stream>>>(
        x, W_enc, hidden, BATCH, HIDDEN_DIM, INPUT_DIM, /*relu=*/0);

    // features(pre-topk) = relu(hidden @ W_exp^T), written straight into d_out
    wmma_gemm_nt<<<dim3(EXPANDED_DIM / 64, BATCH / 64), blk, 0, stream>>>(
        hidden, W_exp, feat, BATCH, EXPANDED_DIM, HIDDEN_DIM, /*relu=*/1);

    // sparsify in place: keep top-K per row
    topk_select<<<dim3(BATCH), dim3(256), 0, stream>>>(feat, TOPK);

    // reconstructed = features @ W_dec^T
    wmma_gemm_nt<<<dim3(INPUT_DIM / 64, BATCH / 64), blk, 0, stream>>>(
        feat, W_dec, recon, BATCH, INPUT_DIM, EXPANDED_DIM, /*relu=*/0);
}